// Qnet_41197326303248
// MI455X (gfx1250) — compile-verified
//
#include <hip/hip_runtime.h>
#include <math.h>

// MI455X (gfx1250) fused MoE-gate kernel.
//
// Bandwidth-bound: ~436 MB of traffic -> ~19 us at 23.3 TB/s; compute is
// ~1.7 GFLOP + ~100M hardware transcendentals. All three K=4 GEMM stages are
// mapped onto V_WMMA_F32_16X16X4_F32 (16x4 A = 2 VGPRs, 4x16 B = 2 VGPRs,
// 16x16 f32 D = 8 VGPRs), with biases folded into the C operand. Layout
// transposes between WMMA stages go through per-wave LDS buffers; the
// elementwise tail (Gumbel noise, sigmoid, hard gate, ReLU, ew2 dots, gating)
// runs per-lane with one row per lane.
// v2: hardware v_tanh_f32 / v_rcp_f32, tile-wide prefetch of A + u1/u2.
// v3: stage-2 LDS scatter split into an unconditional D1 loop plus a single
//     predicated D2 block -> one EXEC toggle instead of eight, letting the
//     stores pair into ds_store_2addr without per-element saveexec churn.

typedef __attribute__((ext_vector_type(2))) float f32x2;
typedef __attribute__((ext_vector_type(4))) float f32x4;
typedef __attribute__((ext_vector_type(8))) float f32x8;

#define WPB       8    // waves per block (256 threads, wave32)
#define H_STRIDE  6    // dwords per row of the h transpose buffer (8B-aligned b64 loads)
#define S2_STRIDE 28   // dwords per row of the stage-2 buffer (16B-aligned b128 loads)
#define GEPS      1e-8f

__device__ __forceinline__ float hw_tanh(float x) {
#if __has_builtin(__builtin_amdgcn_tanhf)
    return __builtin_amdgcn_tanhf(x);          // v_tanh_f32 (CDNA5 trans op)
#else
    float r;
    asm("v_tanh_f32 %0, %1" : "=v"(r) : "v"(x));
    return r;
#endif
}

__device__ __forceinline__ float hw_rcp(float x) {
#if __has_builtin(__builtin_amdgcn_rcpf)
    return __builtin_amdgcn_rcpf(x);           // v_rcp_f32
#else
    float r;
    asm("v_rcp_f32 %0, %1" : "=v"(r) : "v"(x));
    return r;
#endif
}

__device__ __forceinline__ float gumbel_f(float u) {
    // -log(-log(u + eps) + eps), hardware v_log path
    return -__logf(-__logf(u + GEPS) + GEPS);
}

__global__ __launch_bounds__(256) void moe_fused(
    const float* __restrict__ x,
    const float* __restrict__ u1,
    const float* __restrict__ u2,
    const float* __restrict__ fc1_w,   const float* __restrict__ fc1_b,
    const float* __restrict__ picker_w,const float* __restrict__ picker_b,
    const float* __restrict__ ew1,     const float* __restrict__ eb1,
    const float* __restrict__ ew2,     const float* __restrict__ eb2,
    float* __restrict__ outp, float* __restrict__ hardp, float* __restrict__ softp,
    int nTiles)
{
    // Per-wave LDS scratch: h transpose (2 sub-tiles) + stage-2 transpose.
    __shared__ float sH [WPB][2][16 * H_STRIDE];
    __shared__ float sS2[WPB][32 * S2_STRIDE];

    const int lane = threadIdx.x & 31;
    const int wv   = threadIdx.x >> 5;
    const int n    = lane & 15;   // column index within the 16-wide tile
    const int hi   = lane >> 4;   // half-wave: selects K pair {2h, 2h+1}
    const int k0   = 2 * hi;

    // ---------------- Build uniform WMMA B / C operands (once per wave) -------
    // B layout (4x16, K x N): vgpr j, half hi -> K = j + 2*hi, N = lane&15.
    // Stage 1: B = fc1_w^T padded to 16 cols; C = fc1_b broadcast over rows.
    f32x2 Bfc1 = {0.0f, 0.0f};
    if (n < 4) { Bfc1.x = fc1_w[n*4 + k0]; Bfc1.y = fc1_w[n*4 + k0 + 1]; }

    // Stage 2a: cols (4e+j) for experts e=0..3 -> ew1[e][j][k]
    f32x2 B1;
    { int e = n >> 2, j = n & 3;
      B1.x = ew1[e*16 + j*4 + k0];
      B1.y = ew1[e*16 + j*4 + k0 + 1]; }

    // Stage 2b: cols 0..3 = expert 4 (ew1[4][n][k]); cols 4..8 = picker_w rows.
    f32x2 B2 = {0.0f, 0.0f};
    if (n < 4)      { B2.x = ew1[64 + n*4 + k0];       B2.y = ew1[64 + n*4 + k0 + 1]; }
    else if (n < 9) { B2.x = picker_w[(n-4)*4 + k0];   B2.y = picker_w[(n-4)*4 + k0 + 1]; }

    // C operands: bias depends only on the column -> same value in all 8 VGPRs.
    const float cf = (n < 4) ? fc1_b[n] : 0.0f;
    const float c1 = eb1[(n >> 2)*4 + (n & 3)];
    const float c2 = (n < 4) ? eb1[16 + n] : ((n < 9) ? picker_b[n - 4] : 0.0f);
    const f32x8 Cf = {cf,cf,cf,cf,cf,cf,cf,cf};
    const f32x8 C1 = {c1,c1,c1,c1,c1,c1,c1,c1};
    const f32x8 C2 = {c2,c2,c2,c2,c2,c2,c2,c2};

    const int gw = blockIdx.x * WPB + wv;   // global wave id (wave-uniform)
    const int ws = gridDim.x * WPB;

    for (int t = gw; t < nTiles; t += ws) {           // 32 rows per tile
        float* s2 = &sS2[wv][0];
        const long gr = (long)t * 32 + lane;          // tail row owned by this lane

        // ---- Prefetch: both A sub-tiles + all gate noise for this tile ------
        // Issues 12 global loads up front; their latency hides under the
        // WMMA / LDS / transcendental work below.
        f32x2 Apre[2];
        #pragma unroll
        for (int st = 0; st < 2; ++st) {
            const long r0 = (long)t * 32 + st * 16;
            // lane (hi,n) holds x[r0+n][k0 : k0+2] (8B aligned)
            Apre[st] = *(const f32x2*)(x + r0*4 + n*4 + k0);
        }
        float u1v[5], u2v[5];
        #pragma unroll
        for (int e = 0; e < 5; ++e) {
            u1v[e] = u1[gr*5 + e];
            u2v[e] = u2[gr*5 + e];
        }

        #pragma unroll
        for (int st = 0; st < 2; ++st) {              // two 16-row WMMA sub-tiles
            float* hb = &sH[wv][st][0];

            // Stage 1 WMMA: h_pre = X * fc1_w^T + fc1_b   (D cols 0..3 valid)
            f32x8 D = __builtin_amdgcn_wmma_f32_16x16x4_f32(
                false, Apre[st], false, Bfc1, (short)0, Cf, false, false);

            // D-layout -> LDS (16 rows x 4 cols): only lanes with col<4 store.
            if (n < 4) {
                #pragma unroll
                for (int v = 0; v < 8; ++v)
                    hb[(v + 8*hi)*H_STRIDE + n] = D[v];
            }
            // Wave-private cross-lane LDS handoff: hardware keeps DS in order,
            // the asm is a compiler fence + explicit counter wait.
            asm volatile("s_wait_dscnt 0x0" ::: "memory");

            // Reload in A-layout (8B aligned b64), apply hardware tanh.
            f32x2 Ha = *(const f32x2*)(hb + n*H_STRIDE + k0);
            Ha.x = hw_tanh(Ha.x);
            Ha.y = hw_tanh(Ha.y);

            // Stage 2 WMMAs: experts 0..3 | expert 4 + picker logits.
            f32x8 D1 = __builtin_amdgcn_wmma_f32_16x16x4_f32(
                false, Ha, false, B1, (short)0, C1, false, false);
            f32x8 D2 = __builtin_amdgcn_wmma_f32_16x16x4_f32(
                false, Ha, false, B2, (short)0, C2, false, false);

            // Scatter D-layout into the row-major stage-2 buffer:
            // cols 0..15 = eh(e0..e3), 16..19 = eh(e4), 20..24 = logits.
            // D1 stores are unconditional (all 16 columns live): 8 stores that
            // pair into ds_store_2addr with no EXEC churn.
            #pragma unroll
            for (int v = 0; v < 8; ++v) {
                const int m = v + 8*hi + 16*st;
                s2[m*S2_STRIDE + n] = D1[v];
            }
            // D2: only columns 0..8 are live -> ONE predicated block (single
            // EXEC toggle) containing all 8 stores.
            if (n < 9) {
                #pragma unroll
                for (int v = 0; v < 8; ++v) {
                    const int m = v + 8*hi + 16*st;
                    s2[m*S2_STRIDE + 16 + n] = D2[v];
                }
            }
        }
        asm volatile("s_wait_dscnt 0x0" ::: "memory");

        // ---------------- Per-lane tail: one row per lane (32 rows) ----------
        const float* row = s2 + lane * S2_STRIDE;     // 112B stride -> 16B aligned

        f32x4 ehx[5];
        ehx[0] = *(const f32x4*)(row +  0);
        ehx[1] = *(const f32x4*)(row +  4);
        ehx[2] = *(const f32x4*)(row +  8);
        ehx[3] = *(const f32x4*)(row + 12);
        ehx[4] = *(const f32x4*)(row + 16);
        const f32x4 lgv = *(const f32x4*)(row + 20);
        const float lg4 = row[24];

        float o0 = 0.0f, o1 = 0.0f;
        #pragma unroll
        for (int e = 0; e < 5; ++e) {
            const float lg = (e < 4) ? lgv[e] : lg4;
            const float g1 = gumbel_f(u1v[e]);
            const float g2 = gumbel_f(u2v[e]);
            const float z  = lg + (g1 - g2);
            const float s  = hw_rcp(1.0f + __expf(-z));    // sigmoid, temp = 1
            const float hd = (s >= 0.5f) ? 1.0f : 0.0f;    // straight-through hard gate
            softp[gr*5 + e] = s;
            hardp[gr*5 + e] = hd;

            float t0 = eb2[e*2 + 0];
            float t1 = eb2[e*2 + 1];
            #pragma unroll
            for (int j = 0; j < 4; ++j) {
                const float hj = fmaxf(ehx[e][j], 0.0f);   // ReLU(eh)
                t0 = fmaf(hj, ew2[e*8 + j],     t0);       // ew2[e][0][j]
                t1 = fmaf(hj, ew2[e*8 + 4 + j], t1);       // ew2[e][1][j]
            }
            o0 = fmaf(hd, t0, o0);
            o1 = fmaf(hd, t1, o1);
        }
        *(f32x2*)(outp + gr*2) = (f32x2){o0, o1};          // coalesced b64 store
    }
}

extern "C" void kernel_launch(void* const* d_in, const int* in_sizes, int n_in,
                              void* d_out, int out_size, void* d_ws, size_t ws_size,
                              hipStream_t stream) {
    (void)n_in; (void)out_size; (void)d_ws; (void)ws_size;
    const float* x        = (const float*)d_in[0];
    const float* u1       = (const float*)d_in[1];
    const float* u2       = (const float*)d_in[2];
    const float* fc1_w    = (const float*)d_in[3];
    const float* fc1_b    = (const float*)d_in[4];
    const float* picker_w = (const float*)d_in[5];
    const float* picker_b = (const float*)d_in[6];
    const float* ew1      = (const float*)d_in[7];
    const float* eb1      = (const float*)d_in[8];
    const float* ew2      = (const float*)d_in[9];
    const float* eb2      = (const float*)d_in[10];

    const long B = (long)in_sizes[0] / 4;   // x is [B,4]; B = 4194304 (32 | B)
    const int  nTiles = (int)(B / 32);

    float* outp  = (float*)d_out;           // [B,2]
    float* hardp = outp + B * 2;            // [B,5]
    float* softp = outp + B * 7;            // [B,5]

    int blocks = (nTiles + WPB - 1) / WPB;
    if (blocks > 2048) blocks = 2048;       // 16384 waves, ~8 tiles each
    if (blocks < 1)    blocks = 1;

    moe_fused<<<blocks, 256, 0, stream>>>(x, u1, u2, fc1_w, fc1_b,
                                          picker_w, picker_b, ew1, eb1, ew2, eb2,
                                          outp, hardp, softp, nTiles);
}